// TopKGate_22720376996508
// MI455X (gfx1250) — compile-verified
//
#include <hip/hip_runtime.h>
#include <hip/hip_bf16.h>

// ---------------------------------------------------------------------------
// Top-1 MoE gate (TopKGate), MI455X / gfx1250.
//   S=8192 tokens, D=1024, E=64 experts, CAP=128.
// Cost model: output is ~539MB of stores (23us @ 23.3TB/s) -> store bound.
// GEMM (8192x1024x64, 1.07 GFLOP) uses V_WMMA_F32_16X16X4_F32 (full fp32,
// preserves argmax tie behavior), fused with softmax/argmax/partial sums.
// B is staged in LDS with K-pairs interleaved so each WMMA B operand is a
// single ds_load_b64 (no VGPR pack shuffles).
// Sequential cumsum handled by one 64-thread block (1 expert per thread).
// ---------------------------------------------------------------------------

#define S_TOK   8192
#define D_DIM   1024
#define E_EXP   64
#define CAP     128
#define TOK_PER_BLOCK 128           // 8 waves x 16 tokens
#define GEMM_BLOCKS   (S_TOK / TOK_PER_BLOCK)   // 64
#define KB_CHUNK 128                // wg rows staged in LDS per chunk (32KB)

typedef float v2f __attribute__((ext_vector_type(2)));
typedef float v4f __attribute__((ext_vector_type(4)));
typedef float v8f __attribute__((ext_vector_type(8)));

// d_out layout (flat concat, all float):
//   [0]                                   l_aux
//   [1 .. 1+S*E*CAP)                      combine_weights
//   [.. +S*E*CAP)                         dispatch_mask (0/1)
//   [.. +S*E)                             mask1 (0/1)
//   [.. +E)                               exp_counts
//   [.. +S)                               indices1_s
#define OFF_COMB  ((size_t)1)
#define OFF_DISP  (OFF_COMB + (size_t)S_TOK * E_EXP * CAP)
#define OFF_MASK1 (OFF_DISP + (size_t)S_TOK * E_EXP * CAP)
#define OFF_CNT   (OFF_MASK1 + (size_t)S_TOK * E_EXP)
#define OFF_IDX   (OFF_CNT + E_EXP)
#define OUT_TOTAL (OFF_IDX + S_TOK)

// ---------------------------------------------------------------------------
// Kernel 1: zero the whole output with wide stores (the bandwidth hot spot).
// ---------------------------------------------------------------------------
__global__ void k_zero(float* __restrict__ out, size_t n) {
    size_t tid    = (size_t)blockIdx.x * blockDim.x + threadIdx.x;
    size_t stride = (size_t)gridDim.x * blockDim.x;
    size_t n4     = n >> 2;
    v4f z = {0.f, 0.f, 0.f, 0.f};
    v4f* __restrict__ o4 = (v4f*)out;
    for (size_t j = tid; j < n4; j += stride) o4[j] = z;
    size_t tail = n & 3;
    if (tid < tail) out[(n4 << 2) + tid] = 0.f;
}

// ---------------------------------------------------------------------------
// Kernel 2: WMMA GEMM (logits = x @ wg) fused with softmax / argmax /
// per-block expert gate sums. 256 threads = 8 waves; wave w computes a
// 16-token x 64-expert strip via 4 f32 accumulators over K=1024.
//
// LDS B layout ("paired"): element p = (k/2)*64 + n holds the float2
// (wg[k][n], wg[k+1][n]) for even k. WMMA B operand for lane (16h+l),
// tile t, K-step kk is exactly lds_bp[(kk/2 + h)*64 + 16t + l] -> one b64.
// ---------------------------------------------------------------------------
__global__ void __launch_bounds__(256)
k_gemm_gate(const float* __restrict__ x,      // [S, D]
            const float* __restrict__ wg,     // [D, E]
            int*   __restrict__ idx_ws,       // [S]
            float* __restrict__ gmax_ws,      // [S]
            float* __restrict__ me_part) {    // [GEMM_BLOCKS, E]
    __shared__ v2f   lds_bp[(KB_CHUNK / 2) * E_EXP];   // 32 KB: paired wg chunk
    __shared__ float lds_g[TOK_PER_BLOCK * E_EXP];     // 32 KB: logits->gates

    const int wave = threadIdx.x >> 5;
    const int lane = threadIdx.x & 31;
    const int half = lane >> 4;            // selects K pair (2h,2h+1) / M high 8
    const int l    = lane & 15;
    const int m0   = blockIdx.x * TOK_PER_BLOCK + wave * 16;   // token base

    v8f acc0 = {}, acc1 = {}, acc2 = {}, acc3 = {};

    const float* __restrict__ arow = x + (size_t)(m0 + l) * D_DIM;
    float* __restrict__ lds_bf = (float*)lds_bp;

    for (int kb = 0; kb < D_DIM; kb += KB_CHUNK) {
        // Stage wg chunk [kb..kb+128) x 64: coalesced global read, LDS scatter
        // into paired layout (each slot written by exactly one thread).
        const float* __restrict__ wsrc = wg + (size_t)kb * E_EXP;
        for (int i = threadIdx.x; i < KB_CHUNK * E_EXP; i += 256) {
            int kl = i >> 6;          // local k row
            int n  = i & 63;          // expert col
            lds_bf[(((kl >> 1) * E_EXP + n) << 1) | (kl & 1)] = wsrc[i];
        }
        __syncthreads();

        #pragma unroll 4
        for (int kk = 0; kk < KB_CHUNK; kk += 4) {
            // A 16x4 f32 layout: lane (16h+l) holds A[M=l][K=2h], A[M=l][K=2h+1]
            v2f a = *(const v2f*)(arow + kb + kk + 2 * half);
            // B 4x16 f32 layout: lane (16h+l) holds B[K=2h][N=l], B[K=2h+1][N=l]
            const v2f* brow = &lds_bp[((kk >> 1) + half) * E_EXP + l];
            acc0 = __builtin_amdgcn_wmma_f32_16x16x4_f32(false, a, false, brow[0],
                                                         (short)0, acc0, false, false);
            acc1 = __builtin_amdgcn_wmma_f32_16x16x4_f32(false, a, false, brow[16],
                                                         (short)0, acc1, false, false);
            acc2 = __builtin_amdgcn_wmma_f32_16x16x4_f32(false, a, false, brow[32],
                                                         (short)0, acc2, false, false);
            acc3 = __builtin_amdgcn_wmma_f32_16x16x4_f32(false, a, false, brow[48],
                                                         (short)0, acc3, false, false);
        }
        __syncthreads();
    }

    // D layout: VGPR r, lane (16h+l) = D[M=r+8h][N=l]. Spill strip to LDS.
    {
        const int rowbase = wave * 16 + 8 * half;
        #pragma unroll
        for (int r = 0; r < 8; ++r) {
            float* dst = &lds_g[(rowbase + r) * E_EXP + l];
            dst[0]  = acc0[r];
            dst[16] = acc1[r];
            dst[32] = acc2[r];
            dst[48] = acc3[r];
        }
    }
    __syncthreads();

    // Softmax + argmax: one thread per token (threads 0..127).
    if (threadIdx.x < TOK_PER_BLOCK) {
        const int s = blockIdx.x * TOK_PER_BLOCK + threadIdx.x;
        float* row = &lds_g[threadIdx.x * E_EXP];
        float mx = row[0];
        int   am = 0;
        #pragma unroll
        for (int e = 1; e < E_EXP; ++e) {
            float v = row[e];
            if (v > mx) { mx = v; am = e; }   // strict '>' => first max (jnp.argmax)
        }
        float z = 0.f;
        #pragma unroll
        for (int e = 0; e < E_EXP; ++e) {
            float g = __expf(row[e] - mx);
            row[e] = g;
            z += g;
        }
        float inv = 1.f / z;
        #pragma unroll
        for (int e = 0; e < E_EXP; ++e) row[e] *= inv;
        idx_ws[s]  = am;
        gmax_ws[s] = row[am];
    }
    __syncthreads();

    // Deterministic per-block expert gate sums (me partials): thread e sums
    // its column over the 128 tokens in fixed order.
    if (threadIdx.x < E_EXP) {
        float sum = 0.f;
        for (int t = 0; t < TOK_PER_BLOCK; ++t)
            sum += lds_g[t * E_EXP + threadIdx.x];
        me_part[blockIdx.x * E_EXP + threadIdx.x] = sum;
    }
}

// ---------------------------------------------------------------------------
// Kernel 3: sequential cumsum(mask1) scan + exp_counts + l_aux.
// One block, 64 threads: thread e owns expert e, scans all 8192 tokens.
// Deterministic fixed-order reductions only.
// ---------------------------------------------------------------------------
#define SCAN_CHUNK 512
__global__ void __launch_bounds__(64)
k_scan(const int*   __restrict__ idx_ws,   // [S]
       const float* __restrict__ me_part,  // [GEMM_BLOCKS, E]
       int*   __restrict__ loc_ws,         // [S]
       float* __restrict__ out) {          // full output base
    __shared__ int   lds_idx[SCAN_CHUNK];
    __shared__ int   lds_loc[SCAN_CHUNK];
    __shared__ float lds_red[E_EXP];

    const int e = threadIdx.x;
    int count = 0;
    for (int base = 0; base < S_TOK; base += SCAN_CHUNK) {
        for (int i = threadIdx.x; i < SCAN_CHUNK; i += 64)
            lds_idx[i] = idx_ws[base + i];
        __syncthreads();
        for (int i = 0; i < SCAN_CHUNK; ++i) {
            if (lds_idx[i] == e) { lds_loc[i] = count; ++count; }
        }
        __syncthreads();                    // every slot written by its owner
        for (int i = threadIdx.x; i < SCAN_CHUNK; i += 64)
            loc_ws[base + i] = lds_loc[i];
        __syncthreads();
    }

    // exp_counts (float in d_out)
    out[OFF_CNT + e] = (float)count;

    // me[e] = mean over tokens of gates; reduce block partials in fixed order
    float me = 0.f;
    for (int b = 0; b < GEMM_BLOCKS; ++b) me += me_part[b * E_EXP + e];
    me *= (1.f / (float)S_TOK);
    float ce = (float)count * (1.f / (float)S_TOK);
    lds_red[e] = me * ce;
    __syncthreads();
    if (threadIdx.x == 0) {
        float acc = 0.f;
        for (int j = 0; j < E_EXP; ++j) acc += lds_red[j];
        out[0] = acc * (float)E_EXP;        // l_aux
    }
}

// ---------------------------------------------------------------------------
// Kernel 4: scatter the sparse nonzeros into the zeroed output.
// ---------------------------------------------------------------------------
__global__ void k_scatter(const int*   __restrict__ idx_ws,
                          const int*   __restrict__ loc_ws,
                          const float* __restrict__ gmax_ws,
                          float* __restrict__ out) {
    int s = blockIdx.x * blockDim.x + threadIdx.x;
    if (s >= S_TOK) return;
    int   e = idx_ws[s];
    int   c = loc_ws[s];
    float g = gmax_ws[s];
    out[OFF_IDX + s]                       = (float)e;  // indices1_s
    out[OFF_MASK1 + (size_t)s * E_EXP + e] = 1.f;       // mask1 one-hot
    if (c < CAP) {                                      // token kept
        size_t o = (size_t)s * E_EXP * CAP + (size_t)e * CAP + (size_t)c;
        out[OFF_COMB + o] = g;                          // combine_weights
        out[OFF_DISP + o] = (g != 0.f) ? 1.f : 0.f;     // dispatch_mask
    }
}

// ---------------------------------------------------------------------------
extern "C" void kernel_launch(void* const* d_in, const int* in_sizes, int n_in,
                              void* d_out, int out_size, void* d_ws, size_t ws_size,
                              hipStream_t stream) {
    const float* x  = (const float*)d_in[0];   // [4,2048,1024] -> [8192,1024]
    const float* wg = (const float*)d_in[1];   // [1024,64]
    float* out = (float*)d_out;

    // workspace layout
    int*   idx_ws  = (int*)d_ws;                    // 8192 ints
    int*   loc_ws  = idx_ws + S_TOK;                // 8192 ints
    float* gmax_ws = (float*)(loc_ws + S_TOK);      // 8192 floats
    float* me_part = gmax_ws + S_TOK;               // 64*64 floats

    // 1) zero all outputs (store-bandwidth bound: ~539MB -> ~23us @ 23.3TB/s)
    k_zero<<<4096, 256, 0, stream>>>(out, OUT_TOTAL);

    // 2) WMMA GEMM + softmax + argmax + expert partial sums
    k_gemm_gate<<<GEMM_BLOCKS, 256, 0, stream>>>(x, wg, idx_ws, gmax_ws, me_part);

    // 3) sequential per-expert location scan + counts + l_aux
    k_scan<<<1, 64, 0, stream>>>(idx_ws, me_part, loc_ws, out);

    // 4) scatter the <=8192 nonzeros
    k_scatter<<<(S_TOK + 255) / 256, 256, 0, stream>>>(idx_ws, loc_ws, gmax_ws, out);
}